// SoftTripletKLLoss_35588099015101
// MI455X (gfx1250) — compile-verified
//
#include <hip/hip_runtime.h>
#include <hip/hip_bf16.h>
#include <math.h>

// SoftTripletKLLoss for MI455X (gfx1250), algebraically reduced from O(N^3) to O(N^2).
// Gram matrix via V_WMMA_F32_16X16X4_F32 (f32 precision kept because exp() amplifies error).

#define NPTS 4096
#define DDIM 512
#define CCLS 100
#define MARGIN_F 1.0f

typedef __attribute__((ext_vector_type(2))) float v2f;
typedef __attribute__((ext_vector_type(8))) float v8f;

// ---------------- K0: row L2-normalize + x2 ----------------
__global__ __launch_bounds__(128) void stk_rownorm(const float* __restrict__ embs,
                                                   float* __restrict__ e,
                                                   float* __restrict__ x2) {
  const int row = blockIdx.x;
  const int t = threadIdx.x;                  // 128 threads, 4 floats each
  const float4 d = ((const float4*)(embs + (size_t)row * DDIM))[t];
  float ss = d.x * d.x + d.y * d.y + d.z * d.z + d.w * d.w;
  __shared__ float red[128];
  red[t] = ss;
  __syncthreads();
  for (int s = 64; s > 0; s >>= 1) {
    if (t < s) red[t] += red[t + s];
    __syncthreads();
  }
  const float sumsq = red[0];
  const float nrm = fmaxf(sqrtf(sumsq), 1e-12f);
  const float inv = 1.0f / nrm;
  float4 o;
  o.x = d.x * inv; o.y = d.y * inv; o.z = d.z * inv; o.w = d.w * inv;
  ((float4*)(e + (size_t)row * DDIM))[t] = o;
  if (t == 0) x2[row] = sumsq * inv * inv;
}

// ---------------- K1: labels argmax (first max, like jnp.argmax) ----------------
__global__ void stk_argmax(const float* __restrict__ labels, int* __restrict__ idx) {
  const int i = blockIdx.x * blockDim.x + threadIdx.x;
  if (i >= NPTS) return;
  const float* row = labels + (size_t)i * CCLS;
  float best = row[0];
  int bi = 0;
  for (int c = 1; c < CCLS; ++c) {
    const float v = row[c];
    if (v > best) { best = v; bi = c; }
  }
  idx[i] = bi;
}

// ---------------- K2: Gram via WMMA f32 + fused multi-sigma kernel ----------------
// grid = (256, 8), block = 256 (8 waves). Wave w handles row-tile ti=blockIdx.x,
// col-tiles tj0..tj0+3 where tj0 = (blockIdx.y*8 + w)*4.  K = 512 in steps of 4.
__global__ __launch_bounds__(256) void stk_gram_kern(const float* __restrict__ e,
                                                     const float* __restrict__ x2,
                                                     float* __restrict__ kern) {
  const int lane = threadIdx.x & 31;
  const int wave = threadIdx.x >> 5;
  const int half = lane >> 4;   // 0: lanes 0-15, 1: lanes 16-31
  const int id16 = lane & 15;
  const int ti = blockIdx.x;                      // 0..255
  const int tj0 = (blockIdx.y * 8 + wave) * 4;    // 0..252 step 4

  const int arow = ti * 16 + id16;
  const float* aptr = e + (size_t)arow * DDIM + 2 * half;
  const float* bptr0 = e + (size_t)((tj0 + 0) * 16 + id16) * DDIM + 2 * half;
  const float* bptr1 = e + (size_t)((tj0 + 1) * 16 + id16) * DDIM + 2 * half;
  const float* bptr2 = e + (size_t)((tj0 + 2) * 16 + id16) * DDIM + 2 * half;
  const float* bptr3 = e + (size_t)((tj0 + 3) * 16 + id16) * DDIM + 2 * half;

  v8f acc0 = {}, acc1 = {}, acc2 = {}, acc3 = {};
#pragma unroll 8
  for (int k = 0; k < DDIM; k += 4) {
    // A 16x4 f32 fragment: VGPR0 = K + 2*half, VGPR1 = K + 2*half + 1 (M = lane&15)
    const v2f a = *(const v2f*)(aptr + k);
    const v2f b0 = *(const v2f*)(bptr0 + k);
    const v2f b1 = *(const v2f*)(bptr1 + k);
    const v2f b2 = *(const v2f*)(bptr2 + k);
    const v2f b3 = *(const v2f*)(bptr3 + k);
    acc0 = __builtin_amdgcn_wmma_f32_16x16x4_f32(false, a, false, b0, (short)0, acc0, false, false);
    acc1 = __builtin_amdgcn_wmma_f32_16x16x4_f32(false, a, false, b1, (short)0, acc1, false, false);
    acc2 = __builtin_amdgcn_wmma_f32_16x16x4_f32(false, a, false, b2, (short)0, acc2, false, false);
    acc3 = __builtin_amdgcn_wmma_f32_16x16x4_f32(false, a, false, b3, (short)0, acc3, false, false);
  }

  // Epilogue: pdist = x2_i + x2_j - 2 G ; kern = sum_s exp(-pdist/(2*sigma_s))
  const float x2j0 = x2[(tj0 + 0) * 16 + id16];
  const float x2j1 = x2[(tj0 + 1) * 16 + id16];
  const float x2j2 = x2[(tj0 + 2) * 16 + id16];
  const float x2j3 = x2[(tj0 + 3) * 16 + id16];
  const int rbase = ti * 16 + 8 * half;   // C/D layout: VGPR r -> M = r + 8*half
#pragma unroll
  for (int r = 0; r < 8; ++r) {
    const float x2i = x2[rbase + r];
    float* krow = kern + (size_t)(rbase + r) * NPTS;
    const float pd0 = x2i + x2j0 - 2.0f * acc0[r];
    const float pd1 = x2i + x2j1 - 2.0f * acc1[r];
    const float pd2 = x2i + x2j2 - 2.0f * acc2[r];
    const float pd3 = x2i + x2j3 - 2.0f * acc3[r];
    const float k0 = expf(-pd0) + expf(-0.5f * pd0) + expf(-0.25f * pd0) + expf(-0.125f * pd0);
    const float k1 = expf(-pd1) + expf(-0.5f * pd1) + expf(-0.25f * pd1) + expf(-0.125f * pd1);
    const float k2 = expf(-pd2) + expf(-0.5f * pd2) + expf(-0.25f * pd2) + expf(-0.125f * pd2);
    const float k3 = expf(-pd3) + expf(-0.5f * pd3) + expf(-0.25f * pd3) + expf(-0.125f * pd3);
    krow[(tj0 + 0) * 16 + id16] = k0;
    krow[(tj0 + 1) * 16 + id16] = k1;
    krow[(tj0 + 2) * 16 + id16] = k2;
    krow[(tj0 + 3) * 16 + id16] = k3;
  }
}

// ---------------- K3: row sums S and logS ----------------
__global__ __launch_bounds__(256) void stk_rowsum(const float* __restrict__ kern,
                                                  float* __restrict__ S,
                                                  float* __restrict__ logS) {
  const int row = blockIdx.x;
  const int t = threadIdx.x;  // 256
  const float* kr = kern + (size_t)row * NPTS;
  float s = 0.0f;
  for (int j = t; j < NPTS; j += 256) s += kr[j];
  __shared__ float red[256];
  red[t] = s;
  __syncthreads();
  for (int w = 128; w > 0; w >>= 1) {
    if (t < w) red[t] += red[t + w];
    __syncthreads();
  }
  if (t == 0) {
    const float v = red[0];
    S[row] = v;
    logS[row] = logf(v);
  }
}

// ---------------- K4: per-class stats (deterministic, no atomics) ----------------
__global__ __launch_bounds__(128) void stk_cls(const int* __restrict__ idx,
                                               const float* __restrict__ logS,
                                               int* __restrict__ cnt,
                                               float* __restrict__ clsLogS,
                                               float* __restrict__ sumLogS) {
  const int t = threadIdx.x;  // 128, one class each (c < 100)
  if (t < CCLS) {
    int n = 0;
    float s = 0.0f;
    for (int j = 0; j < NPTS; ++j) {
      if (idx[j] == t) { ++n; s += logS[j]; }
    }
    cnt[t] = n;
    clsLogS[t] = s;
  }
  __syncthreads();
  if (t == 0) {
    float s = 0.0f;
    for (int c = 0; c < CCLS; ++c) s += clsLogS[c];
    sumLogS[0] = s;  // every point belongs to exactly one class
  }
}

// ---------------- K5: class-scatter M[c,k] = sum_{j in c} log(kern[j,k]), T[k] ----------------
// 32 blocks x 128 threads; each thread owns one column (no cross-thread LDS sharing).
__global__ __launch_bounds__(128) void stk_classM(const float* __restrict__ kern,
                                                  const int* __restrict__ idx,
                                                  float* __restrict__ Mlog,
                                                  float* __restrict__ Tlog) {
  __shared__ float Ml[CCLS * 128];  // 51.2 KB of the 320 KB WGP LDS
  const int t = threadIdx.x;
  const int col = blockIdx.x * 128 + t;
  for (int c = 0; c < CCLS; ++c) Ml[c * 128 + t] = 0.0f;
  float tl = 0.0f;
  for (int j = 0; j < NPTS; ++j) {
    const float v = logf(kern[(size_t)j * NPTS + col]);
    Ml[idx[j] * 128 + t] += v;  // idx[j] uniform across wave -> conflict-free banks
    tl += v;
  }
  for (int c = 0; c < CCLS; ++c) Mlog[(size_t)c * NPTS + col] = Ml[c * 128 + t];
  Tlog[col] = tl;
}

// ---------------- K6: per-row loss ----------------
__global__ __launch_bounds__(128) void stk_rowloss(const float* __restrict__ kern,
                                                   const int* __restrict__ idx,
                                                   const float* __restrict__ S,
                                                   const float* __restrict__ logS,
                                                   const float* __restrict__ Mlog,
                                                   const float* __restrict__ Tlog,
                                                   const int* __restrict__ cnt,
                                                   const float* __restrict__ clsLogS,
                                                   const float* __restrict__ sumLogS,
                                                   float* __restrict__ rowloss) {
  const int i = blockIdx.x;
  const int t = threadIdx.x;  // 128
  const int c = idx[i];
  const float* kr = kern + (size_t)i * NPTS;
  const float* mr = Mlog + (size_t)c * NPTS;
  float dM = 0.0f, dT = 0.0f, dH = 0.0f;
  for (int k = t; k < NPTS; k += 128) {
    const float kv = kr[k];
    dM += kv * mr[k];
    dT += kv * Tlog[k];
    dH += kv * logf(kv);
  }
  __shared__ float rm[128], rt[128], rh[128];
  rm[t] = dM; rt[t] = dT; rh[t] = dH;
  __syncthreads();
  for (int w = 64; w > 0; w >>= 1) {
    if (t < w) { rm[t] += rm[t + w]; rt[t] += rt[t + w]; rh[t] += rh[t + w]; }
    __syncthreads();
  }
  if (t == 0) {
    const float invS = 1.0f / S[i];
    const float H = rh[0] * invS - logS[i];                 // H_i = sum_k P log P (row i)
    const float PiM = rm[0] * invS;                         // sum_k P[i,k] Mlog[c,k]
    const float PiT = rt[0] * invS;                         // sum_k P[i,k] Tlog[k]
    // sum_j (2*pos-1)*kl[i,j] = (2*npos-N)*H_i - 2*(PiM - clsLogS[c]) + (PiT - sumLogS)
    const float term = (2.0f * (float)cnt[c] - (float)NPTS) * H
                       - 2.0f * (PiM - clsLogS[c]) + (PiT - sumLogS[0]);
    const float rl = term * (1.0f / (float)NPTS) + MARGIN_F;
    rowloss[i] = fmaxf(rl, 0.0f);
  }
}

// ---------------- K7: final mean ----------------
__global__ __launch_bounds__(256) void stk_final(const float* __restrict__ rowloss,
                                                 float* __restrict__ out) {
  const int t = threadIdx.x;  // 256
  float s = 0.0f;
  for (int j = t; j < NPTS; j += 256) s += rowloss[j];
  __shared__ float red[256];
  red[t] = s;
  __syncthreads();
  for (int w = 128; w > 0; w >>= 1) {
    if (t < w) red[t] += red[t + w];
    __syncthreads();
  }
  if (t == 0) out[0] = red[0] * (1.0f / (float)NPTS);
}

extern "C" void kernel_launch(void* const* d_in, const int* in_sizes, int n_in,
                              void* d_out, int out_size, void* d_ws, size_t ws_size,
                              hipStream_t stream) {
  (void)in_sizes; (void)n_in; (void)out_size; (void)ws_size;
  const float* embs   = (const float*)d_in[0];  // [4096, 512] f32
  const float* labels = (const float*)d_in[1];  // [4096, 100] f32 one-hot

  char* ws = (char*)d_ws;
  size_t off = 0;
  float* e       = (float*)(ws + off); off += (size_t)NPTS * DDIM * sizeof(float);  // 8 MB
  float* kern    = (float*)(ws + off); off += (size_t)NPTS * NPTS * sizeof(float);  // 64 MB
  float* Mlog    = (float*)(ws + off); off += (size_t)CCLS * NPTS * sizeof(float);  // 1.6 MB
  float* x2      = (float*)(ws + off); off += (size_t)NPTS * sizeof(float);
  float* S       = (float*)(ws + off); off += (size_t)NPTS * sizeof(float);
  float* logS    = (float*)(ws + off); off += (size_t)NPTS * sizeof(float);
  float* Tlog    = (float*)(ws + off); off += (size_t)NPTS * sizeof(float);
  float* rowloss = (float*)(ws + off); off += (size_t)NPTS * sizeof(float);
  int*   idx     = (int*)  (ws + off); off += (size_t)NPTS * sizeof(int);
  int*   cnt     = (int*)  (ws + off); off += 512;
  float* clsLogS = (float*)(ws + off); off += 512;
  float* sumLogS = (float*)(ws + off); off += 256;

  stk_rownorm<<<NPTS, 128, 0, stream>>>(embs, e, x2);
  stk_argmax<<<NPTS / 128, 128, 0, stream>>>(labels, idx);
  {
    dim3 grid(NPTS / 16, 8, 1);  // 256 row-tiles x (8 * 8 waves) = 64 col-strips of width 64
    stk_gram_kern<<<grid, 256, 0, stream>>>(e, x2, kern);
  }
  stk_rowsum<<<NPTS, 256, 0, stream>>>(kern, S, logS);
  stk_cls<<<1, 128, 0, stream>>>(idx, logS, cnt, clsLogS, sumLogS);
  stk_classM<<<NPTS / 128, 128, 0, stream>>>(kern, idx, Mlog, Tlog);
  stk_rowloss<<<NPTS, 128, 0, stream>>>(kern, idx, S, logS, Mlog, Tlog, cnt, clsLogS,
                                        sumLogS, rowloss);
  stk_final<<<1, 256, 0, stream>>>(rowloss, (float*)d_out);
}